// User_embedding_53772990546508
// MI455X (gfx1250) — compile-verified
//
#include <hip/hip_runtime.h>
#include <hip/hip_bf16.h>
#include <stdint.h>

#define BATCH   8
#define EDGES_M 100000
#define EDGES_T 100000
#define EDGES   200000          // per batch after concat (day 0 only)
#define NUSER   50000
#define NLOC    10000
#define DIM     200
#define UPB     8               // users per block (8 wave32 waves)
#define UBLK    (NUSER / UPB)   // 6250 blocks per batch

typedef int v4i __attribute__((ext_vector_type(4)));
typedef int v8i __attribute__((ext_vector_type(8)));

// ---------------------------------------------------------------- init
__global__ void k_init(int* __restrict__ counts, float* __restrict__ avg_sum,
                       int* __restrict__ nwe) {
    int i = blockIdx.x * 256 + threadIdx.x;
    if (i < BATCH * NUSER) counts[i] = 0;
    if (i < BATCH * DIM)   avg_sum[i] = 0.f;
    if (i < BATCH)         nwe[i] = 0;
}

// ---------------------------------------------------------------- edge decode
__device__ __forceinline__ const int* edge_ptr(const int* __restrict__ mob,
                                               const int* __restrict__ txt,
                                               int b, int e) {
    return (e < EDGES_M) ? (mob + ((size_t)b * EDGES_M + e) * 2)
                         : (txt + ((size_t)b * EDGES_T + (e - EDGES_M)) * 2);
}

// ---------------------------------------------------------------- count per user
__global__ void k_count(const int* __restrict__ mob, const int* __restrict__ txt,
                        const int* __restrict__ su, int* __restrict__ counts) {
    int idx = blockIdx.x * 256 + threadIdx.x;
    if (idx >= BATCH * EDGES) return;
    int b = idx / EDGES, e = idx - b * EDGES;
    const int* p = edge_ptr(mob, txt, b, e);
    int u = su[p[0]];
    atomicAdd(&counts[b * NUSER + u], 1);
}

// ---------------------------------------------------------------- exclusive scan (1 block / batch)
__global__ void k_scan(const int* __restrict__ counts, int* __restrict__ offsets,
                       int* __restrict__ cursors, int* __restrict__ nwe) {
    __shared__ int sd[1024];
    __shared__ int carry, snwe;
    int b = blockIdx.x, tid = threadIdx.x;
    if (tid == 0) { carry = 0; snwe = 0; }
    __syncthreads();
    const int* c = counts + b * NUSER;
    const int nchunk = (NUSER + 1023) / 1024;
    for (int chunk = 0; chunk < nchunk; ++chunk) {
        int idx = chunk * 1024 + tid;
        int v = (idx < NUSER) ? c[idx] : 0;
        if (v > 0) atomicAdd(&snwe, 1);
        sd[tid] = v;
        __syncthreads();
        for (int off = 1; off < 1024; off <<= 1) {
            int t = (tid >= off) ? sd[tid - off] : 0;
            __syncthreads();
            sd[tid] += t;
            __syncthreads();
        }
        int excl = carry + sd[tid] - v;
        if (idx < NUSER) {
            offsets[b * NUSER + idx] = excl;
            cursors[b * NUSER + idx] = excl;
        }
        __syncthreads();
        if (tid == 1023) carry += sd[1023];
        __syncthreads();
    }
    if (tid == 0) nwe[b] = snwe;
}

// ---------------------------------------------------------------- CSR fill (location idx per slot)
__global__ void k_fill(const int* __restrict__ mob, const int* __restrict__ txt,
                       const int* __restrict__ su, const int* __restrict__ sl,
                       int* __restrict__ cursors, int* __restrict__ elist) {
    int idx = blockIdx.x * 256 + threadIdx.x;
    if (idx >= BATCH * EDGES) return;
    int b = idx / EDGES, e = idx - b * EDGES;
    const int* p = edge_ptr(mob, txt, b, e);
    int u = su[p[0]];
    int l = sl[p[1]];
    int pos = atomicAdd(&cursors[b * NUSER + u], 1);
    elist[(size_t)b * EDGES + pos] = l;
}

// ---------------------------------------------------------------- per-user mean (1 wave / user)
__global__ void k_accum(const float* __restrict__ xloc, const int* __restrict__ counts,
                        const int* __restrict__ offsets, const int* __restrict__ elist,
                        float* __restrict__ out, float* __restrict__ avg_sum) {
    __shared__ float sred[DIM];
    int b     = blockIdx.x / UBLK;
    int ubase = (blockIdx.x % UBLK) * UPB;
    int tid = threadIdx.x;
    int wave = tid >> 5, lane = tid & 31;
    for (int i = tid; i < DIM; i += 256) sred[i] = 0.f;
    __syncthreads();

    int u = ubase + wave;
    int cnt = counts[b * NUSER + u];
    if (cnt > 0) {
        float acc[7];
#pragma unroll
        for (int k = 0; k < 7; ++k) acc[k] = 0.f;
        long base = (long)b * EDGES + offsets[b * NUSER + u];
        for (int eb = 0; eb < cnt; eb += 32) {
            int eid = 0;
            if (eb + lane < cnt) eid = elist[base + eb + lane];
            int n = min(32, cnt - eb);
            for (int j = 0; j < n; ++j) {
                int l = __shfl(eid, j, 32);
                const float* row = xloc + (long)l * DIM;
                if (j + 1 < n) {  // gfx1250 global_prefetch_b8 for next gather row
                    int ln = __shfl(eid, j + 1, 32);
                    __builtin_prefetch(xloc + (long)ln * DIM + lane * 8, 0, 1);
                }
#pragma unroll
                for (int k = 0; k < 7; ++k) {
                    int d = lane + 32 * k;
                    if (d < DIM) acc[k] += row[d];
                }
            }
        }
        float inv = 1.f / (float)cnt;
        float* op = out + ((long)b * NUSER + u) * DIM;
#pragma unroll
        for (int k = 0; k < 7; ++k) {
            int d = lane + 32 * k;
            if (d < DIM) {
                float m = acc[k] * inv;
                op[d] = m;
                atomicAdd(&sred[d], m);   // LDS ds_add_f32
            }
        }
    }
    __syncthreads();
    for (int i = tid; i < DIM; i += 256) {
        float v = sred[i];
        if (v != 0.f) atomicAdd(&avg_sum[b * DIM + i], v);  // one global add / block / dim
    }
}

// ---------------------------------------------------------------- broadcast avg to edge-less users
__global__ void k_bcast(const int* __restrict__ counts, const float* __restrict__ avg_sum,
                        const int* __restrict__ nwe, float* __restrict__ out) {
    __shared__ float s_avg[DIM];
    int b     = blockIdx.x / UBLK;
    int ubase = (blockIdx.x % UBLK) * UPB;
    int tid = threadIdx.x;
    int wave = tid >> 5, lane = tid & 31;

    if (wave == 0) {
        // Tensor Data Mover: DMA 200 f32 (1 x 200 tile, 4B elements) global -> LDS.
        // 2-group D# per CDNA5 ISA: group2/3 omitted (<=2D tensor).
        uint64_t ga  = (uint64_t)(uintptr_t)(avg_sum + (size_t)b * DIM);
        uint32_t lds = (uint32_t)(uintptr_t)(__attribute__((address_space(3))) float*)s_avg;
        v4i g0;
        g0.x = 1;                                   // count=1, user descriptor
        g0.y = (int)lds;                            // lds_addr (bytes)
        g0.z = (int)(uint32_t)(ga & 0xFFFFFFFFu);   // global_addr[31:0]
        g0.w = (int)(uint32_t)(((ga >> 32) & 0x01FFFFFFu) | 0x80000000u); // addr[56:32] | type=2
        v8i g1;
        g1.s0 = (2 << 16);          // workgroup_mask=0, data_size=2 (4B), no pad/iterate/atomic
        g1.s1 = (DIM & 0xFFFF) << 16;   // tensor_dim0[15:0] (bits 79:48 -> hi half of word1)
        g1.s2 = (1 << 16);              // tensor_dim0[31:16]=0 | tensor_dim1=1
        g1.s3 = (DIM & 0xFFFF) << 16;   // tensor_dim1[31:16]=0 | tile_dim0=200
        g1.s4 = 1;                      // tile_dim1=1, tile_dim2=0
        g1.s5 = DIM;                    // tensor_dim0_stride[31:0]
        g1.s6 = 0;                      // stride[47:32]=0 | tensor_dim1_stride[15:0]=0
        g1.s7 = 0;                      // tensor_dim1_stride[47:16]=0
        asm volatile("tensor_load_to_lds %0, %1" :: "s"(g0), "s"(g1) : "memory");
#if __has_builtin(__builtin_amdgcn_s_wait_tensorcnt)
        __builtin_amdgcn_s_wait_tensorcnt(0);
#else
        asm volatile("s_wait_tensorcnt 0x0" ::: "memory");
#endif
    }
    __syncthreads();

    int n = nwe[b];
    if (n < 1) n = 1;
    float inv = 1.f / (float)n;
    int u = ubase + wave;
    if (counts[b * NUSER + u] == 0) {
        float* op = out + ((long)b * NUSER + u) * DIM;
#pragma unroll
        for (int k = 0; k < 7; ++k) {
            int d = lane + 32 * k;
            if (d < DIM) op[d] = s_avg[d] * inv;
        }
    }
}

// ---------------------------------------------------------------- launch
extern "C" void kernel_launch(void* const* d_in, const int* in_sizes, int n_in,
                              void* d_out, int out_size, void* d_ws, size_t ws_size,
                              hipStream_t stream) {
    (void)in_sizes; (void)n_in; (void)out_size; (void)ws_size;
    const float* xloc = (const float*)d_in[0];   // [10000, 200]
    const int*   mob  = (const int*)d_in[1];     // [8,1,100000,2]
    const int*   txt  = (const int*)d_in[2];     // [8,1,100000,2]
    const int*   su   = (const int*)d_in[3];     // [50000]
    const int*   sl   = (const int*)d_in[4];     // [10000]
    float* out = (float*)d_out;                  // [8, 50000, 200]

    // workspace layout (ints): counts | offsets | cursors | edge_list | avg_sum | nwe  (~11.2 MB)
    int* wsI      = (int*)d_ws;
    int* counts   = wsI;
    int* offsets  = wsI + BATCH * NUSER;
    int* cursors  = wsI + 2 * BATCH * NUSER;
    int* elist    = wsI + 3 * BATCH * NUSER;
    float* avgsum = (float*)(wsI + 3 * BATCH * NUSER + BATCH * EDGES);
    int* nwe      = (int*)(avgsum + BATCH * DIM);

    int iblocks = (BATCH * NUSER + 255) / 256;
    int eblocks = (BATCH * EDGES + 255) / 256;

    k_init <<<iblocks, 256, 0, stream>>>(counts, avgsum, nwe);
    k_count<<<eblocks, 256, 0, stream>>>(mob, txt, su, counts);
    k_scan <<<BATCH, 1024, 0, stream>>>(counts, offsets, cursors, nwe);
    k_fill <<<eblocks, 256, 0, stream>>>(mob, txt, su, sl, cursors, elist);
    k_accum<<<BATCH * UBLK, 256, 0, stream>>>(xloc, counts, offsets, elist, out, avgsum);
    k_bcast<<<BATCH * UBLK, 256, 0, stream>>>(counts, avgsum, nwe, out);
}